// Block_8383776162052
// MI455X (gfx1250) — compile-verified
//
#include <hip/hip_runtime.h>
#include <math.h>

typedef __attribute__((ext_vector_type(2))) float v2f;
typedef __attribute__((ext_vector_type(8))) float v8f;

#define V_N   196608
#define DEG_N 9
#define C_N   64
#define K_CH  8
#define EPS_V 1e-5f
#define NBLK_STATS 512
#define VC ((size_t)V_N * C_N)

// ---------------- BatchNorm statistics: deterministic two-pass ----------------
__global__ void bn_stats_partial(const float* __restrict__ x,
                                 float* __restrict__ psum, float* __restrict__ psq) {
    __shared__ float ssum[256];
    __shared__ float ssq[256];
    int t   = threadIdx.x;
    int c   = t & 63;
    int sub = t >> 6;                      // 0..3
    const int rows = V_N / NBLK_STATS;     // 384
    int v0  = blockIdx.x * rows;
    float s = 0.f, q = 0.f;
    for (int r = sub; r < rows; r += 4) {
        float val = x[(size_t)(v0 + r) * C_N + c];
        s += val;
        q += val * val;
    }
    ssum[t] = s; ssq[t] = q;
    __syncthreads();
    if (sub == 0) {
        float S = ssum[c] + ssum[c + 64] + ssum[c + 128] + ssum[c + 192];
        float Q = ssq[c]  + ssq[c + 64]  + ssq[c + 128]  + ssq[c + 192];
        psum[blockIdx.x * 64 + c] = S;
        psq[blockIdx.x * 64 + c]  = Q;
    }
}

__global__ void bn_finalize(const float* __restrict__ psum, const float* __restrict__ psq,
                            float* __restrict__ mean, float* __restrict__ inv) {
    int c = threadIdx.x;                    // 0..63
    float S = 0.f, Q = 0.f;
    for (int b = 0; b < NBLK_STATS; ++b) {  // fixed order -> deterministic
        S += psum[b * 64 + c];
        Q += psq[b * 64 + c];
    }
    float m   = S / (float)V_N;
    float var = Q / (float)V_N - m * m;
    mean[c] = m;
    inv[c]  = rsqrtf(var + EPS_V);
}

// ---------------- normalize + Mish -> x0 ----------------
__global__ void norm_mish(const float* __restrict__ x,
                          const float* __restrict__ mean, const float* __restrict__ inv,
                          const float* __restrict__ g, const float* __restrict__ bt,
                          float* __restrict__ x0) {
    size_t i = (size_t)blockIdx.x * 256 + threadIdx.x;
    int c = (int)(i & 63);
    float h  = (x[i] - mean[c]) * inv[c] * g[c] + bt[c];
    float sp = (h > 20.f) ? h : log1pf(expf(h));   // softplus, overflow-safe
    x0[i] = h * tanhf(sp);
}

// ---------------- init output with bias ----------------
__global__ void init_out(const float* __restrict__ bias, float* __restrict__ out) {
    size_t i = (size_t)blockIdx.x * 256 + threadIdx.x;
    out[i] = bias[i & 63];
}

// ---------------- Chebyshev SpMV: xout = alpha * L@xin + beta * xprev ----------------
__global__ void cheb_spmv(const float* __restrict__ xin, const float* __restrict__ xprev,
                          const int* __restrict__ cols, const float* __restrict__ vals,
                          float* __restrict__ xout, float alpha, float beta) {
    int t = threadIdx.x;
    int c = t & 63;
    int r = t >> 6;                         // 0..3
    int v = blockIdx.x * 4 + r;
    const int*   cp = cols + (size_t)v * DEG_N;
    const float* vp = vals + (size_t)v * DEG_N;
    float acc = 0.f;
#pragma unroll
    for (int d = 0; d < DEG_N; ++d) {
        int col = cp[d];
        float w = vp[d];
        acc += w * xin[(size_t)col * C_N + c];
    }
    xout[(size_t)v * C_N + c] = alpha * acc + beta * xprev[(size_t)v * C_N + c];
}

// ---------------- dense accumulate: out += X @ Wk  (M=V, K=64, N=64) via WMMA fp32 ----
// One wave32 computes a 16(row) x 64(col) tile: 16 K-steps x 4 N-tiles of
// v_wmma_f32_16x16x4_f32. Wk is staged in LDS PRE-SWIZZLED into the exact
// per-lane B-fragment layout, so each fragment is a single aligned ds_load_b64.
//
// LDS layout: for K-row i, N-col o:
//   fragment f = i>>2, tile nt = o>>4, lane = (o&15) + 16*((i&3)>>1), comp = i&1
//   lw[(f*4 + nt)*64 + lane*2 + comp]
__global__ void gemm_acc(const float* __restrict__ X, const float* __restrict__ Wk,
                         float* __restrict__ out) {
    __shared__ float lw[64 * 64];
    int t = threadIdx.x;
    for (int idx = t; idx < 64 * 64; idx += 256) {
        int i  = idx >> 6;          // K row
        int o  = idx & 63;          // N col
        int f  = i >> 2;
        int nt = o >> 4;
        int ln = (o & 15) + ((i & 2) << 3);   // 16*((i&3)>>1)
        int cm = i & 1;
        lw[(((f << 2) + nt) << 6) + (ln << 1) + cm] = Wk[idx];
    }
    __syncthreads();

    int wave = t >> 5;            // 0..7
    int lane = t & 31;
    int vbase = blockIdx.x * 128 + wave * 16;

    int m  = lane & 15;           // A-matrix row (M)
    int kr = (lane >> 4) * 2;     // K sub-offset within fragment: 0 or 2
    int n  = lane & 15;           // C/D column within 16-wide tile

    v8f acc0 = {}, acc1 = {}, acc2 = {}, acc3 = {};
    const float* arow = X + (size_t)(vbase + m) * C_N;
    const float* bl   = lw + (lane << 1);

#pragma unroll
    for (int kk = 0; kk < 64; kk += 4) {
        // A fragment (16x4 fp32): lane L<16 holds K = kk,kk+1; lane>=16 holds kk+2,kk+3
        v2f a = *(const v2f*)(arow + kk + kr);
        // B fragments: contiguous v2f per lane, tiles 256B apart
        int base = kk << 6;                       // (kk/4)*4*64
        v2f b0 = *(const v2f*)(bl + base);
        v2f b1 = *(const v2f*)(bl + base + 64);
        v2f b2 = *(const v2f*)(bl + base + 128);
        v2f b3 = *(const v2f*)(bl + base + 192);

        acc0 = __builtin_amdgcn_wmma_f32_16x16x4_f32(false, a, false, b0, (short)0, acc0, false, false);
        acc1 = __builtin_amdgcn_wmma_f32_16x16x4_f32(false, a, false, b1, (short)0, acc1, false, false);
        acc2 = __builtin_amdgcn_wmma_f32_16x16x4_f32(false, a, false, b2, (short)0, acc2, false, false);
        acc3 = __builtin_amdgcn_wmma_f32_16x16x4_f32(false, a, false, b3, (short)0, acc3, false, false);
    }

    // D layout: VGPR j -> row (j + 8*(lane>>4)), col = lane&15 within each 16-wide tile
    int rowoff = (lane >> 4) * 8;
#pragma unroll
    for (int j = 0; j < 8; ++j) {
        size_t o = (size_t)(vbase + rowoff + j) * C_N;
        out[o +  0 + n] += acc0[j];
        out[o + 16 + n] += acc1[j];
        out[o + 32 + n] += acc2[j];
        out[o + 48 + n] += acc3[j];
    }
}

// ---------------- host-side orchestration ----------------
extern "C" void kernel_launch(void* const* d_in, const int* in_sizes, int n_in,
                              void* d_out, int out_size, void* d_ws, size_t ws_size,
                              hipStream_t stream) {
    (void)in_sizes; (void)n_in; (void)out_size; (void)ws_size;
    const float* x      = (const float*)d_in[0];
    // d_in[1] = lap_rows: structurally v*9..v*9+8, used implicitly
    const int*   cols   = (const int*)d_in[2];
    const float* vals   = (const float*)d_in[3];
    const float* gamma  = (const float*)d_in[4];
    const float* beta   = (const float*)d_in[5];
    const float* weight = (const float*)d_in[6];  // [8][64][64]
    const float* bias   = (const float*)d_in[7];
    float* out = (float*)d_out;

    float* wsf  = (float*)d_ws;
    float* mean = wsf;                            // 64
    float* inv  = wsf + 64;                       // 64
    float* psum = wsf + 128;                      // 512*64
    float* psq  = psum + NBLK_STATS * 64;         // 512*64
    float* xa   = psq + NBLK_STATS * 64;          // V*C each
    float* xb   = xa + VC;
    float* xc   = xb + VC;

    // BN stats (deterministic two-pass)
    bn_stats_partial<<<NBLK_STATS, 256, 0, stream>>>(x, psum, psq);
    bn_finalize<<<1, 64, 0, stream>>>(psum, psq, mean, inv);

    // h = BN(x); x0 = mish(h)
    norm_mish<<<(int)(VC / 256), 256, 0, stream>>>(x, mean, inv, gamma, beta, xa);

    // out = bias; out += x0 @ W[0]
    init_out<<<(int)(VC / 256), 256, 0, stream>>>(bias, out);
    gemm_acc<<<V_N / 128, 256, 0, stream>>>(xa, weight + 0 * 4096, out);

    // x1 = L @ x0 ; out += x1 @ W[1]
    cheb_spmv<<<V_N / 4, 256, 0, stream>>>(xa, xa, cols, vals, xb, 1.f, 0.f);
    gemm_acc<<<V_N / 128, 256, 0, stream>>>(xb, weight + 1 * 4096, out);

    // x_k = 2 L x_{k-1} - x_{k-2} ; out += x_k @ W[k]
    float* prev = xa; float* cur = xb; float* nxt = xc;
    for (int k = 2; k < K_CH; ++k) {
        cheb_spmv<<<V_N / 4, 256, 0, stream>>>(cur, prev, cols, vals, nxt, 2.f, -1.f);
        gemm_acc<<<V_N / 128, 256, 0, stream>>>(nxt, weight + k * 4096, out);
        float* tmp = prev; prev = cur; cur = nxt; nxt = tmp;
    }
}